// ImplicitPhysicsAttention_37744172597596
// MI455X (gfx1250) — compile-verified
//
#include <hip/hip_runtime.h>
#include <hip/hip_bf16.h>

typedef __attribute__((ext_vector_type(2))) float v2f;
typedef __attribute__((ext_vector_type(8))) float v8f;

__device__ inline v8f wmma_k4(v2f a, v2f b, v8f c) {
  // D = A(16x4 f32) x B(4x16 f32) + C(16x16 f32)
  return __builtin_amdgcn_wmma_f32_16x16x4_f32(false, a, false, b, (short)0, c, false, false);
}

__device__ inline float wsum32(float v) {
  #pragma unroll
  for (int o = 16; o > 0; o >>= 1) v += __shfl_xor(v, o, 32);
  return v;
}

// ---------------------------------------------------------------------------
// Stage 1: per-(b, channel) border statistics for the algebraically-collapsed
// conv+meanpool. x viewed as [b][32][32][256]; channel i.
// stats[wid][9] = {T, R0, R31, C0, C31, x00, x0_31, x31_0, x31_31}
// ---------------------------------------------------------------------------
__global__ void stats_kernel(const float* __restrict__ x, float* __restrict__ stats) {
  int wave = threadIdx.x >> 5, lane = threadIdx.x & 31;
  int wid = blockIdx.x * (blockDim.x >> 5) + wave;   // 0..2047
  int b = wid >> 8, i = wid & 255;
  int r = lane;
  const float* base = x + ((size_t)(b * 1024 + r * 32)) * 256 + i;
  float rowsum = 0.f, first = 0.f, last = 0.f;
  for (int c = 0; c < 32; ++c) {
    float v = base[(size_t)c * 256];
    rowsum += v;
    if (c == 0) first = v;
    if (c == 31) last = v;
  }
  float R0     = __shfl(rowsum, 0, 32);
  float R31    = __shfl(rowsum, 31, 32);
  float x00    = __shfl(first, 0, 32);
  float x0_31  = __shfl(last, 0, 32);
  float x31_0  = __shfl(first, 31, 32);
  float x31_31 = __shfl(last, 31, 32);
  float T   = wsum32(rowsum);
  float C0  = wsum32(first);
  float C31 = wsum32(last);
  if (lane == 0) {
    float* st = stats + (size_t)wid * 9;
    st[0] = T;  st[1] = R0;  st[2] = R31; st[3] = C0; st[4] = C31;
    st[5] = x00; st[6] = x0_31; st[7] = x31_0; st[8] = x31_31;
  }
}

// ---------------------------------------------------------------------------
// Stage 2: pooled -> phy -> alpha[b*8+h] = scale * scat / (atten + 1e-6)
// ---------------------------------------------------------------------------
__global__ void phy_kernel(const float* __restrict__ stats,
                           const float* __restrict__ conv_w,
                           const float* __restrict__ conv_b,
                           const float* __restrict__ phy_w,
                           const float* __restrict__ phy_b,
                           float* __restrict__ alpha_out) {
  __shared__ float pooled_s[8][32];
  __shared__ float atten_s[8][8];
  __shared__ float scat_s[8][8];
  int t = threadIdx.x;
  int b = t >> 5, o = t & 31;
  float acc = 0.f;
  for (int i = 0; i < 256; ++i) {
    const float* st = stats + (size_t)(b * 256 + i) * 9;
    float T = st[0], R0 = st[1], R31 = st[2], C0 = st[3], C31 = st[4];
    float x00 = st[5], x0_31 = st[6], x31_0 = st[7], x31_31 = st[8];
    const float* w = conv_w + (size_t)(o * 256 + i) * 9;
    acc += w[0] * (T - R31 - C31 + x31_31);   // dh=-1, dw=-1
    acc += w[1] * (T - R31);                  // dh=-1, dw= 0
    acc += w[2] * (T - R31 - C0 + x31_0);     // dh=-1, dw=+1
    acc += w[3] * (T - C31);                  // dh= 0, dw=-1
    acc += w[4] * T;                          // dh= 0, dw= 0
    acc += w[5] * (T - C0);                   // dh= 0, dw=+1
    acc += w[6] * (T - R0 - C31 + x0_31);     // dh=+1, dw=-1
    acc += w[7] * (T - R0);                   // dh=+1, dw= 0
    acc += w[8] * (T - R0 - C0 + x00);        // dh=+1, dw=+1
  }
  pooled_s[b][o] = acc * (1.f / 1024.f) + conv_b[o];
  __syncthreads();
  if (t < 128) {
    int bb = t >> 4, j = t & 15;
    float ph = phy_b[j];
    for (int o2 = 0; o2 < 32; ++o2) ph += pooled_s[bb][o2] * phy_w[j * 32 + o2];
    if (j < 8) atten_s[bb][j] = ph; else scat_s[bb][j - 8] = ph;
  }
  __syncthreads();
  if (t < 64) {
    int bb = t >> 3, hh = t & 7;
    alpha_out[t] = 0.17677669529663687f * scat_s[bb][hh] / (atten_s[bb][hh] + 1e-6f);
  }
}

// ---------------------------------------------------------------------------
// Stage 3: flash-style physics attention per (b,h):
//   out = softmax(alpha * q @ q^T) @ q,  q = x[b,:,h*32:(h+1)*32] (1024x32)
// One wave owns 16 query rows. Score tile computed TRANSPOSED (M=keys,
// N=rows) so each row's stats reduce with a single xor-16 shuffle.
// Two independent WMMA accumulator chains for the S tile to pipeline the
// f32 WMMA D->C RAW hazard.
// ---------------------------------------------------------------------------
#define FA_WAVES 4
__global__ __launch_bounds__(128) void flash_phys_kernel(const float* __restrict__ x,
                                                         const float* __restrict__ alpha_buf,
                                                         float* __restrict__ attn_out) {
  int wave = threadIdx.x >> 5, lane = threadIdx.x & 31;
  int half = lane >> 4, l16 = lane & 15;
  int bh = blockIdx.y;
  int b = bh >> 3, h = bh & 7;
  int rowbase = (blockIdx.x * FA_WAVES + wave) << 4;
  float alpha = alpha_buf[bh];
  const float* qbase = x + (size_t)b * 1024 * 256 + h * 32;

  // Query-row fragment in B layout: B[k=hd, n=row]; per kb: k = kb*4 + {0,1} + 2*half
  v2f qB[8];
  #pragma unroll
  for (int kb = 0; kb < 8; ++kb)
    qB[kb] = *(const v2f*)(qbase + (size_t)(rowbase + l16) * 256 + kb * 4 + 2 * half);

  float mrow = -1e30f, lrow = 0.f;
  v8f O0 = {}, O1 = {};

  for (int c0 = 0; c0 < 1024; c0 += 16) {
    // S^T tile: A = key-cols fragment (M=col, K=hd), B = qB; two chains
    v8f sa = {}, sb = {};
    #pragma unroll
    for (int kb = 0; kb < 8; kb += 2) {
      v2f a0 = *(const v2f*)(qbase + (size_t)(c0 + l16) * 256 + kb * 4 + 2 * half);
      v2f a1 = *(const v2f*)(qbase + (size_t)(c0 + l16) * 256 + (kb + 1) * 4 + 2 * half);
      sa = wmma_k4(a0, qB[kb], sa);
      sb = wmma_k4(a1, qB[kb + 1], sb);
    }
    // lane l: row = l16, keys (v + 8*half) for v=0..7
    float p[8];
    float pm = -1e30f;
    #pragma unroll
    for (int v = 0; v < 8; ++v) {
      float sv = (sa[v] + sb[v]) * alpha;
      p[v] = sv;
      pm = fmaxf(pm, sv);
    }
    pm = fmaxf(pm, __shfl_xor(pm, 16, 32));
    float m_new = fmaxf(mrow, pm);
    float corr = __expf(mrow - m_new);
    float ls = 0.f;
    #pragma unroll
    for (int v = 0; v < 8; ++v) { p[v] = __expf(p[v] - m_new); ls += p[v]; }
    ls += __shfl_xor(ls, 16, 32);
    lrow = lrow * corr + ls;
    mrow = m_new;
    // rescale O: O-reg v corresponds to row (v + 8*half); corr lives in lane==row
    #pragma unroll
    for (int v = 0; v < 8; ++v) {
      float cb = __shfl(corr, (lane < 16) ? v : (v + 8), 32);
      O0[v] *= cb;
      O1[v] *= cb;
    }
    // O += P(16 rows x 16 keys) @ Qc(16 keys x 32): build P A-frags by shuffle
    #pragma unroll
    for (int kb2 = 0; kb2 < 16; kb2 += 4) {
      v2f af;
      #pragma unroll
      for (int j = 0; j < 2; ++j) {
        const int k0 = kb2 + j;        // key if half==0
        const int k1 = kb2 + j + 2;    // key if half==1
        float v0 = __shfl(p[k0 & 7], l16 + 16 * (k0 >> 3), 32);
        float v1 = __shfl(p[k1 & 7], l16 + 16 * (k1 >> 3), 32);
        af[j] = half ? v1 : v0;
      }
      const float* colp = qbase + (size_t)(c0 + kb2 + 2 * half) * 256;
      {
        v2f bf; bf.x = colp[l16]; bf.y = colp[256 + l16];
        O0 = wmma_k4(af, bf, O0);
      }
      {
        v2f bf; bf.x = colp[16 + l16]; bf.y = colp[256 + 16 + l16];
        O1 = wmma_k4(af, bf, O1);
      }
    }
  }
  float linv = 1.f / lrow;
  #pragma unroll
  for (int v = 0; v < 8; ++v) {
    float lb = __shfl(linv, (lane < 16) ? v : (v + 8), 32);
    size_t row = (size_t)(b * 1024 + rowbase + v + 8 * half);
    attn_out[row * 256 + h * 32 + l16]      = O0[v] * lb;
    attn_out[row * 256 + h * 32 + 16 + l16] = O1[v] * lb;
  }
}

// ---------------------------------------------------------------------------
// Generic fp32 WMMA GEMM: C[M x N] = A[M x K] @ W[N x K]^T + bias
// One 16x64 tile per wave: 4 independent WMMA accumulator chains (pipelines
// the f32 WMMA RAW hazard) and 4x reuse of each A fragment.
// Requires N % 64 == 0, M % 16 == 0, K % 4 == 0.
// ---------------------------------------------------------------------------
__global__ __launch_bounds__(256) void gemm_wmma(const float* __restrict__ A, int lda,
                                                 const float* __restrict__ W,
                                                 const float* __restrict__ bias,
                                                 float* __restrict__ C, int ldc,
                                                 int M, int N, int K) {
  int wave = threadIdx.x >> 5, lane = threadIdx.x & 31;
  int half = lane >> 4, l16 = lane & 15;
  int tile = blockIdx.x * (blockDim.x >> 5) + wave;
  int ntn = N >> 6;                  // 64-wide column groups
  int tm = (tile / ntn) << 4;
  int tn = (tile % ntn) << 6;
  if (tm >= M) return;
  const float* arow  = A + (size_t)(tm + l16) * lda + 2 * half;   // A: M = l16
  const float* wrow0 = W + (size_t)(tn + l16) * K + 2 * half;     // B: N = l16
  const float* wrow1 = wrow0 + (size_t)16 * K;
  const float* wrow2 = wrow0 + (size_t)32 * K;
  const float* wrow3 = wrow0 + (size_t)48 * K;
  __builtin_prefetch(arow, 0, 1);      // global_prefetch_b8
  __builtin_prefetch(wrow0, 0, 1);
  __builtin_prefetch(wrow1, 0, 1);
  __builtin_prefetch(wrow2, 0, 1);
  __builtin_prefetch(wrow3, 0, 1);
  v8f acc0 = {}, acc1 = {}, acc2 = {}, acc3 = {};
  #pragma unroll 4
  for (int k = 0; k < K; k += 4) {
    v2f af = *(const v2f*)(arow + k);
    acc0 = wmma_k4(af, *(const v2f*)(wrow0 + k), acc0);
    acc1 = wmma_k4(af, *(const v2f*)(wrow1 + k), acc1);
    acc2 = wmma_k4(af, *(const v2f*)(wrow2 + k), acc2);
    acc3 = wmma_k4(af, *(const v2f*)(wrow3 + k), acc3);
  }
  float bv0 = bias[tn + l16];
  float bv1 = bias[tn + 16 + l16];
  float bv2 = bias[tn + 32 + l16];
  float bv3 = bias[tn + 48 + l16];
  #pragma unroll
  for (int v = 0; v < 8; ++v) {
    size_t m = (size_t)(tm + v + 8 * half);
    C[m * ldc + tn + l16]      = acc0[v] + bv0;
    C[m * ldc + tn + 16 + l16] = acc1[v] + bv1;
    C[m * ldc + tn + 32 + l16] = acc2[v] + bv2;
    C[m * ldc + tn + 48 + l16] = acc3[v] + bv3;
  }
}

// ---------------------------------------------------------------------------
// Stage 5: env attention. One thread per (z, b): 8+64 keys, online softmax.
//   scores use head slice z%8 ; env VALUES use head slice z//1024 (reference!)
// ---------------------------------------------------------------------------
__global__ __launch_bounds__(256) void env_attn_kernel(const float* __restrict__ qe,
                                                       const float* __restrict__ ke,
                                                       const float* __restrict__ ve,
                                                       const float* __restrict__ envk,
                                                       const float* __restrict__ lambda_p,
                                                       float* __restrict__ oute) {
  int tid = blockIdx.x * blockDim.x + threadIdx.x;  // 65536
  int bq = tid & 7;
  int z = tid >> 3;
  int n_idx = z >> 3;
  int h_idx = z & 7;
  int h_v = z >> 10;
  const float SCALE = 0.17677669529663687f;  // 1/sqrt(32)
  float lam = *lambda_p;

  float q[32];
  const float* qp = qe + ((size_t)(bq * 1024 + n_idx) * 256 + h_idx * 32);
  #pragma unroll
  for (int d = 0; d < 32; ++d) q[d] = qp[d];

  float m = -1e30f, l = 0.f;
  float acc[32];
  #pragma unroll
  for (int d = 0; d < 32; ++d) acc[d] = 0.f;

  for (int c = 0; c < 8; ++c) {
    const float* kp = ke + ((size_t)(c * 1024 + n_idx) * 256 + h_idx * 32);
    const float* vp = ve + ((size_t)(c * 1024 + n_idx) * 256 + h_idx * 32);
    float s = 0.f;
    #pragma unroll
    for (int d = 0; d < 32; ++d) s += q[d] * kp[d];
    s *= SCALE;
    float mn = fmaxf(m, s);
    float corr = __expf(m - mn);
    float p = __expf(s - mn);
    l = l * corr + p;
    #pragma unroll
    for (int d = 0; d < 32; ++d) acc[d] = acc[d] * corr + p * vp[d];
    m = mn;
  }
  for (int e = 0; e < 64; ++e) {
    const float* kp = envk + (size_t)e * 256 + h_idx * 32;
    const float* vp = envk + (size_t)e * 256 + h_v * 32;
    float s = 0.f;
    #pragma unroll
    for (int d = 0; d < 32; ++d) s += q[d] * kp[d];
    s *= lam * SCALE;
    float mn = fmaxf(m, s);
    float corr = __expf(m - mn);
    float p = __expf(s - mn);
    l = l * corr + p;
    #pragma unroll
    for (int d = 0; d < 32; ++d) acc[d] = acc[d] * corr + p * vp[d];
    m = mn;
  }
  float inv = 1.f / l;
  float* op = oute + ((size_t)(bq * 1024 + n_idx) * 256 + h_idx * 32);
  #pragma unroll
  for (int d = 0; d < 32; ++d) op[d] = acc[d] * inv;
}

// ---------------------------------------------------------------------------
// Stage 7: g = sigmoid(glin); out = g*cat[:, :256] + (1-g)*cat[:, 256:]
// ---------------------------------------------------------------------------
__global__ void gate_blend_kernel(const float* __restrict__ cat,
                                  const float* __restrict__ glin,
                                  float* __restrict__ outp) {
  size_t idx = (size_t)blockIdx.x * blockDim.x + threadIdx.x;  // 2097152
  size_t row = idx >> 8;
  int col = (int)(idx & 255);
  float g = 1.f / (1.f + __expf(-glin[idx]));
  float o1 = cat[row * 512 + col];
  float o2 = cat[row * 512 + 256 + col];
  outp[idx] = g * o1 + (1.f - g) * o2;
}

extern "C" void kernel_launch(void* const* d_in, const int* in_sizes, int n_in,
                              void* d_out, int out_size, void* d_ws, size_t ws_size,
                              hipStream_t stream) {
  const float* x          = (const float*)d_in[0];
  const float* conv_w     = (const float*)d_in[1];
  const float* conv_b     = (const float*)d_in[2];
  const float* phy_w      = (const float*)d_in[3];
  const float* phy_b      = (const float*)d_in[4];
  const float* to_out_w   = (const float*)d_in[5];
  const float* to_out_b   = (const float*)d_in[6];
  const float* env_keys   = (const float*)d_in[7];
  const float* env_proj_w = (const float*)d_in[8];
  const float* env_proj_b = (const float*)d_in[9];
  const float* lambda_p   = (const float*)d_in[10];
  const float* q_w        = (const float*)d_in[11];
  const float* q_b        = (const float*)d_in[12];
  const float* k_w        = (const float*)d_in[13];
  const float* k_b        = (const float*)d_in[14];
  const float* v_w        = (const float*)d_in[15];
  const float* v_b        = (const float*)d_in[16];
  const float* out_proj_w = (const float*)d_in[17];
  const float* out_proj_b = (const float*)d_in[18];
  const float* gate_w     = (const float*)d_in[19];
  const float* gate_b     = (const float*)d_in[20];
  float* out = (float*)d_out;

  float* ws = (float*)d_ws;
  float* stats    = ws;                     // 18432
  float* alpha    = ws + 18432;             // 64
  float* qe       = ws + 32768;             // 2097152
  float* ke       = qe + 2097152;
  float* ve       = ke + 2097152;
  float* envk     = ve + 2097152;           // 16384
  float* attn_out = envk + 16384;           // 2097152
  float* oute     = attn_out + 2097152;     // 2097152
  float* cat      = oute + 2097152;         // 4194304 (8192 x 512)
  float* glin     = cat + 4194304;          // 2097152

  stats_kernel<<<256, 256, 0, stream>>>(x, stats);
  phy_kernel<<<1, 256, 0, stream>>>(stats, conv_w, conv_b, phy_w, phy_b, alpha);
  flash_phys_kernel<<<dim3(16, 64), 128, 0, stream>>>(x, alpha, attn_out);
  // out = attn_out @ to_out_w^T + b  -> cat[:, 0:256]   tiles = 512*4 = 2048
  gemm_wmma<<<256, 256, 0, stream>>>(attn_out, 256, to_out_w, to_out_b, cat, 512, 8192, 256, 256);
  // q/k/v projections
  gemm_wmma<<<256, 256, 0, stream>>>(x, 256, q_w, q_b, qe, 256, 8192, 256, 256);
  gemm_wmma<<<256, 256, 0, stream>>>(x, 256, k_w, k_b, ke, 256, 8192, 256, 256);
  gemm_wmma<<<256, 256, 0, stream>>>(x, 256, v_w, v_b, ve, 256, 8192, 256, 256);
  // env_k = env_keys @ env_proj_w^T + b   tiles = 4*4 = 16
  gemm_wmma<<<2, 256, 0, stream>>>(env_keys, 256, env_proj_w, env_proj_b, envk, 256, 64, 256, 256);
  env_attn_kernel<<<256, 256, 0, stream>>>(qe, ke, ve, envk, lambda_p, oute);
  // out_e = oute @ out_proj_w^T + b -> cat[:, 256:512]
  gemm_wmma<<<256, 256, 0, stream>>>(oute, 256, out_proj_w, out_proj_b, cat + 256, 512, 8192, 256, 256);
  // gate logits: cat(8192x512) @ gate_w^T(512x256) + b
  gemm_wmma<<<256, 256, 0, stream>>>(cat, 512, gate_w, gate_b, glin, 256, 8192, 256, 512);
  gate_blend_kernel<<<8192, 256, 0, stream>>>(cat, glin, out);
}